// Resample2d_65180423684748
// MI455X (gfx1250) — compile-verified
//
#include <hip/hip_runtime.h>
#include <hip/hip_bf16.h>

// Resample2d (bilinear warp), MI455X / gfx1250.
// B=8, D=64, H=384, W=512, all f32. Memory-bound: ~402MB img read (L2-resident
// per-batch slabs, 192MB L2) + ~402MB streamed output + 12.6MB flow => ~35us
// floor at 23.3 TB/s. Thread == output pixel, loop over 64 channels so the
// bilinear setup lives in registers and is amortized 64x.
// CDNA5 paths: flow row staged via global_load_async_to_lds_b32 (ASYNCcnt) +
// s_wait_asynccnt; output uses non-temporal stores (th:NT) so streaming writes
// don't evict reusable taps from L2.
// Tap optimization: both taps of a row are loaded as one b64 at column
// bx = min(x0, W-2); the x==W-1 clamp case is folded into per-pixel pair
// weights (exact, since clamped wx==0 there), halving gather requests.

#define BDIM 8
#define DDIM 64
#define HDIM 384
#define WDIM 512
#define TPB  256

#if defined(__has_builtin)
#if __has_builtin(__builtin_amdgcn_global_load_async_to_lds_b32) && \
    __has_builtin(__builtin_amdgcn_s_wait_asynccnt)
#define USE_ASYNC_LDS 1
#endif
#endif

// Async-to-LDS builtin on this toolchain: (global int*, shared int*, imm, imm).
typedef __attribute__((address_space(1))) int gint_t;
typedef __attribute__((address_space(3))) int lint_t;

// 8-byte vector load at 4-byte alignment (gfx1250 handles unaligned VMEM).
typedef float f2_t __attribute__((ext_vector_type(2), aligned(4)));

__global__ __launch_bounds__(TPB) void Resample2d_65180423684748_kernel(
    const float* __restrict__ img,   // (B, D, H, W)
    const float* __restrict__ flow,  // (B, 2, H, W)
    float* __restrict__ out)         // (B, D, H, W)
{
    const int tid = threadIdx.x;
    const int w = blockIdx.x * TPB + tid;
    const int h = blockIdx.y;
    const int b = blockIdx.z;

    const int plane = HDIM * WDIM;
    const float* uptr = flow + ((size_t)b * 2) * plane + (size_t)h * WDIM + w;
    const float* vptr = uptr + plane;

    float u, v;
#if USE_ASYNC_LDS
    __shared__ float s_flow[2 * TPB];
    // Async DMA of the flow row chunk straight into LDS (tracked by ASYNCcnt).
    __builtin_amdgcn_global_load_async_to_lds_b32(
        (gint_t*)uptr, (lint_t*)&s_flow[tid], 0, 0);
    __builtin_amdgcn_global_load_async_to_lds_b32(
        (gint_t*)vptr, (lint_t*)&s_flow[TPB + tid], 0, 0);
    __builtin_amdgcn_s_wait_asynccnt(0);
    __syncthreads();
    u = s_flow[tid];
    v = s_flow[TPB + tid];
#else
    u = *uptr;
    v = *vptr;
#endif

    // Bilinear setup (once per pixel, amortized over 64 channels).
    float fx = (float)w + u;
    float fy = (float)h + v;
    fx = fminf(fmaxf(fx, 0.0f), (float)(WDIM - 1));
    fy = fminf(fmaxf(fy, 0.0f), (float)(HDIM - 1));
    const float x0f = floorf(fx);
    const float y0f = floorf(fy);
    const int x0 = (int)x0f;
    const int y0 = (int)y0f;
    const int y1 = min(y0 + 1, HDIM - 1);
    const float wx = fx - x0f;   // == 0 when x0 == W-1 (fx clamped)
    const float wy = fy - y0f;   // == 0 when y0 == H-1 (fy clamped)
    const float wx0 = 1.0f - wx;
    const float wy0 = 1.0f - wy;
    const float w00 = wy0 * wx0;
    const float w01 = wy0 * wx;
    const float w10 = wy * wx0;
    const float w11 = wy * wx;

    // Pairwise-load weight fold: load float2 at bx = min(x0, W-2).
    // Normal case: (pair.x, pair.y) = (x0, x0+1) -> weights (w?0, w?1).
    // Edge x0==W-1: pair covers (W-2, W-1), desired tap is pair.y with the
    // full row weight (w01==w11==0 there), so weights become (0, w00)/(0, w10).
    const bool xedge = (x0 >= WDIM - 1);
    const int bx = xedge ? (WDIM - 2) : x0;
    const float a0 = xedge ? 0.0f : w00;
    const float a1 = xedge ? w00 : w01;
    const float b0 = xedge ? 0.0f : w10;
    const float b1 = xedge ? w10 : w11;

    const size_t oA = (size_t)y0 * WDIM + bx;  // top tap pair
    const size_t oB = (size_t)y1 * WDIM + bx;  // bottom tap pair

    const float* ibase = img + (size_t)b * DDIM * plane;
    float* obase = out + (size_t)b * DDIM * plane + (size_t)h * WDIM + w;

    // 64 channels: 2 coalesced b64 tap loads + 4 FMA-class ops + 1 NT store.
#pragma unroll 8
    for (int d = 0; d < DDIM; ++d) {
        const float* p = ibase + (size_t)d * plane;
        const f2_t pa = *(const f2_t*)(p + oA);
        const f2_t pb = *(const f2_t*)(p + oB);
        float acc = a0 * pa.x;
        acc = fmaf(a1, pa.y, acc);
        acc = fmaf(b0, pb.x, acc);
        acc = fmaf(b1, pb.y, acc);
        __builtin_nontemporal_store(acc, obase + (size_t)d * plane);
    }
}

extern "C" void kernel_launch(void* const* d_in, const int* in_sizes, int n_in,
                              void* d_out, int out_size, void* d_ws, size_t ws_size,
                              hipStream_t stream) {
    (void)in_sizes; (void)n_in; (void)out_size; (void)d_ws; (void)ws_size;
    const float* img  = (const float*)d_in[0];
    const float* flow = (const float*)d_in[1];
    float* out = (float*)d_out;
    dim3 grid(WDIM / TPB, HDIM, BDIM);  // (2, 384, 8)
    dim3 block(TPB, 1, 1);              // 8 x wave32
    Resample2d_65180423684748_kernel<<<grid, block, 0, stream>>>(img, flow, out);
}